// SGConvNet_79611513798694
// MI455X (gfx1250) — compile-verified
//
#include <hip/hip_runtime.h>
#include <math.h>

typedef __attribute__((ext_vector_type(2))) float v2f;
typedef __attribute__((ext_vector_type(8))) float v8f;

#define TB 256

// ---------------------------------------------------------------- utilities
__global__ void zero_i32(int* __restrict__ p, int n) {
  int i = blockIdx.x * blockDim.x + threadIdx.x;
  if (i < n) p[i] = 0;
}

__global__ void count_edges(const int* __restrict__ dst, int* __restrict__ count, int e) {
  int i = blockIdx.x * blockDim.x + threadIdx.x;
  if (i < e) atomicAdd(&count[dst[i]], 1);
}

// Single-workgroup scan over N=50000 counts -> exclusive rowStart[N+1].
__global__ void scan_counts(const int* __restrict__ count, int* __restrict__ rowStart, int n) {
  __shared__ int part[1024];
  const int t = threadIdx.x;
  const int chunk = (n + 1023) >> 10;
  int b = t * chunk; if (b > n) b = n;
  int e = b + chunk; if (e > n) e = n;
  int s = 0;
  for (int i = b; i < e; ++i) s += count[i];
  part[t] = s;
  __syncthreads();
  for (int off = 1; off < 1024; off <<= 1) {
    int v = (t >= off) ? part[t - off] : 0;
    __syncthreads();
    part[t] += v;
    __syncthreads();
  }
  int base = (t == 0) ? 0 : part[t - 1];
  for (int i = b; i < e; ++i) { rowStart[i] = base; base += count[i]; }
  if (t == 1023) rowStart[n] = part[1023];
}

// dis[i] = 1/sqrt(deg) with deg = in-degree + self loop; also zero the fill cursors.
__global__ void dis_and_fill(const int* __restrict__ count, float* __restrict__ dis,
                             int* __restrict__ fill, int n) {
  int i = blockIdx.x * blockDim.x + threadIdx.x;
  if (i < n) {
    dis[i] = 1.0f / sqrtf((float)(count[i] + 1));
    fill[i] = 0;
  }
}

// CSR scatter: col[p]=src, val[p]=dis[src]*dis[dst]
__global__ void scatter_edges(const int* __restrict__ src, const int* __restrict__ dst,
                              const int* __restrict__ rowStart, int* __restrict__ fill,
                              int* __restrict__ col, float* __restrict__ val,
                              const float* __restrict__ dis, int e) {
  int i = blockIdx.x * blockDim.x + threadIdx.x;
  if (i < e) {
    int d = dst[i], s = src[i];
    int p = rowStart[d] + atomicAdd(&fill[d], 1);
    col[p] = s;
    val[p] = dis[s] * dis[d];
  }
}

// -------------------------------------------------- WMMA fp32 GEMM (Ncols=128)
// D[M,128] = relu(A[M,K] @ B[K,128] + bias).  One wave owns one 16x16 tile of D;
// the 8 waves of a block cover the 8 column tiles of one 16-row strip.
// V_WMMA_F32_16X16X4_F32 operand layout (ISA 7.12.2):
//   A 16x4  : lanes 0-15 -> M=lane, V0:K=0 V1:K=1 ; lanes 16-31 -> V0:K=2 V1:K=3
//   B 4x16  : lanes 0-15 -> rows K=0(V0),K=1(V1) ; lanes 16-31 -> K=2(V0),K=3(V1)
//   C/D     : VGPR i, lanes 0-15 -> M=i, lanes 16-31 -> M=8+i, N=lane&15
template <int K>
__global__ __launch_bounds__(256) void gemm128_bias_relu(
    const float* __restrict__ A, const float* __restrict__ B,
    const float* __restrict__ bias, float* __restrict__ D, int M) {
  const int lane = threadIdx.x & 31;
  const int wv   = threadIdx.x >> 5;     // 0..7 -> column tile
  const int m0   = blockIdx.x * 16;
  const int n0   = wv * 16;
  const int half = lane >> 4;            // 0 or 1
  const int l    = lane & 15;

  int arow = m0 + l; if (arow >= M) arow = M - 1;   // clamp (grids are exact anyway)
  const float* Ap = A + (size_t)arow * K + half * 2;
  const float* Bp = B + (size_t)(half * 2) * 128 + n0 + l;

  v8f c = {0.f, 0.f, 0.f, 0.f, 0.f, 0.f, 0.f, 0.f};
#pragma unroll
  for (int k0 = 0; k0 < K; k0 += 4) {
    v2f a = *(const v2f*)(Ap + k0);            // A[row][k0+2*half .. +1], 8B aligned
    v2f b;
    b.x = Bp[(size_t)(k0 + 0) * 128];          // B[k0+2*half  ][n0+l]
    b.y = Bp[(size_t)(k0 + 1) * 128];          // B[k0+2*half+1][n0+l]
    c = __builtin_amdgcn_wmma_f32_16x16x4_f32(
        /*neg_a=*/false, a, /*neg_b=*/false, b,
        /*c_mod=*/(short)0, c, /*reuse_a=*/false, /*reuse_b=*/false);
  }

  const float bv = bias[n0 + l];
#pragma unroll
  for (int i = 0; i < 8; ++i) {
    int row = m0 + half * 8 + i;
    if (row < M) {
      float v = c[i] + bv;
      D[(size_t)row * 128 + n0 + l] = v > 0.f ? v : 0.f;
    }
  }
}

// ------------------------------------------- one hop of normalized propagation
// hout[i] = dis[i]^2 * hin[i] + sum_{p in row i} val[p] * hin[col[p]]
// One wave per node; each lane owns 4 contiguous features (128 = 32*4),
// gathers are float4 (global_load_b128) and stay L2-resident (25.6MB << 192MB).
__global__ __launch_bounds__(256) void sgc_aggregate(
    const float* __restrict__ hin, float* __restrict__ hout,
    const int* __restrict__ rowStart, const int* __restrict__ col,
    const float* __restrict__ val, const float* __restrict__ dis, int n) {
  const int node = (blockIdx.x * blockDim.x + threadIdx.x) >> 5;
  const int lane = threadIdx.x & 31;
  if (node >= n) return;
  const size_t f0 = (size_t)lane * 4;

  const float d  = dis[node];
  const float sw = d * d;
  float4 hv = *(const float4*)(hin + (size_t)node * 128 + f0);
  float ax = sw * hv.x, ay = sw * hv.y, az = sw * hv.z, aw = sw * hv.w;

  const int s = rowStart[node];
  const int e = rowStart[node + 1];
  for (int p = s; p < e; ++p) {
    int   cidx = col[p];
    float w    = val[p];
    float4 g = *(const float4*)(hin + (size_t)cidx * 128 + f0);
    ax = fmaf(w, g.x, ax);
    ay = fmaf(w, g.y, ay);
    az = fmaf(w, g.z, az);
    aw = fmaf(w, g.w, aw);
  }
  float4 o; o.x = ax; o.y = ay; o.z = az; o.w = aw;
  *(float4*)(hout + (size_t)node * 128 + f0) = o;
}

// ------------------------------------------------ final projection H=128 -> 1
__global__ __launch_bounds__(256) void lin3_relu(
    const float* __restrict__ h, const float* __restrict__ W3,
    const float* __restrict__ b3, float* __restrict__ out, int n) {
  const int node = (blockIdx.x * blockDim.x + threadIdx.x) >> 5;
  const int lane = threadIdx.x & 31;
  if (node >= n) return;
  const size_t f0 = (size_t)lane * 4;
  float4 hv = *(const float4*)(h + (size_t)node * 128 + f0);
  float4 wv = *(const float4*)(W3 + f0);
  float s = hv.x * wv.x + hv.y * wv.y + hv.z * wv.z + hv.w * wv.w;
#pragma unroll
  for (int off = 16; off > 0; off >>= 1) s += __shfl_down(s, off, 32);
  if (lane == 0) {
    float v = s + b3[0];
    out[node] = v > 0.f ? v : 0.f;
  }
}

// ---------------------------------------------------------------- launcher
extern "C" void kernel_launch(void* const* d_in, const int* in_sizes, int n_in,
                              void* d_out, int out_size, void* d_ws, size_t ws_size,
                              hipStream_t stream) {
  const float* x   = (const float*)d_in[0];
  const int*   ei  = (const int*)  d_in[1];   // [2,E] flat: src = ei, dst = ei+E
  const float* W0  = (const float*)d_in[2];
  const float* b0  = (const float*)d_in[3];
  const float* Wc1 = (const float*)d_in[4];
  const float* bc1 = (const float*)d_in[5];
  const float* Wc2 = (const float*)d_in[6];
  const float* bc2 = (const float*)d_in[7];
  const float* W1  = (const float*)d_in[8];
  const float* b1  = (const float*)d_in[9];
  const float* W2  = (const float*)d_in[10];
  const float* b2  = (const float*)d_in[11];
  const float* W3  = (const float*)d_in[12];
  const float* b3  = (const float*)d_in[13];
  float* out = (float*)d_out;

  const int F = in_sizes[2] / 128;      // W0 is [F_IN,128]  -> 64
  const int N = in_sizes[0] / F;        // x  is [N,F_IN]    -> 50000
  const int E = in_sizes[1] / 2;        // edge_index [2,E]  -> 1.6M

  // carve workspace (256B aligned slices)
  char* wp = (char*)d_ws;
  auto carve = [&](size_t bytes) -> void* {
    void* r = (void*)wp;
    wp += (bytes + 255) & ~(size_t)255;
    return r;
  };
  int*   count    = (int*)  carve((size_t)N * 4);
  int*   fill     = (int*)  carve((size_t)N * 4);
  int*   rowStart = (int*)  carve((size_t)(N + 1) * 4);
  float* dis      = (float*)carve((size_t)N * 4);
  int*   col      = (int*)  carve((size_t)E * 4);
  float* val      = (float*)carve((size_t)E * 4);
  float* hA       = (float*)carve((size_t)N * 128 * 4);
  float* hB       = (float*)carve((size_t)N * 128 * 4);

  const int gN = (N + TB - 1) / TB;
  const int gE = (E + TB - 1) / TB;
  const int gW = (N + 7) / 8;           // wave-per-node kernels, 8 waves/block
  const int gM = (N + 15) / 16;         // 16-row GEMM strips (exact for N=50000)

  // ---- graph structure (rebuilt identically every call)
  zero_i32     <<<gN, TB, 0, stream>>>(count, N);
  count_edges  <<<gE, TB, 0, stream>>>(ei + E, count, E);
  scan_counts  <<<1, 1024, 0, stream>>>(count, rowStart, N);
  dis_and_fill <<<gN, TB, 0, stream>>>(count, dis, fill, N);
  scatter_edges<<<gE, TB, 0, stream>>>(ei, ei + E, rowStart, fill, col, val, dis, E);

  // ---- h = relu(x @ W0 + b0)
  if (F == 64)
    gemm128_bias_relu<64> <<<gM, 256, 0, stream>>>(x, W0, b0, hA, N);
  else
    gemm128_bias_relu<128><<<gM, 256, 0, stream>>>(x, W0, b0, hA, N);

  // ---- SGConv1: 3 hops then GEMM
  sgc_aggregate<<<gW, 256, 0, stream>>>(hA, hB, rowStart, col, val, dis, N);
  sgc_aggregate<<<gW, 256, 0, stream>>>(hB, hA, rowStart, col, val, dis, N);
  sgc_aggregate<<<gW, 256, 0, stream>>>(hA, hB, rowStart, col, val, dis, N);
  gemm128_bias_relu<128><<<gM, 256, 0, stream>>>(hB, Wc1, bc1, hA, N);

  // ---- SGConv2: 3 hops then GEMM
  sgc_aggregate<<<gW, 256, 0, stream>>>(hA, hB, rowStart, col, val, dis, N);
  sgc_aggregate<<<gW, 256, 0, stream>>>(hB, hA, rowStart, col, val, dis, N);
  sgc_aggregate<<<gW, 256, 0, stream>>>(hA, hB, rowStart, col, val, dis, N);
  gemm128_bias_relu<128><<<gM, 256, 0, stream>>>(hB, Wc2, bc2, hA, N);

  // ---- lin1, lin2
  gemm128_bias_relu<128><<<gM, 256, 0, stream>>>(hA, W1, b1, hB, N);
  gemm128_bias_relu<128><<<gM, 256, 0, stream>>>(hB, W2, b2, hA, N);

  // ---- lin3 + relu -> out [N,1]
  lin3_relu<<<gW, 256, 0, stream>>>(hA, W3, b3, out, N);
}